// S2Convolution_31705448579123
// MI455X (gfx1250) — compile-verified
//
#include <hip/hip_runtime.h>
#include <math.h>

typedef float v2f __attribute__((ext_vector_type(2)));
typedef float v8f __attribute__((ext_vector_type(8)));

#define NSPEC3 5456
#define TWO_PI 6.28318530717958647692f

// cumulative offsets of (2l+1)^2 blocks, l = 0..15 (+ total)
__constant__ int c_off[17] = {0, 1, 10, 35, 84, 165, 286, 455, 680, 969,
                              1330, 1771, 2300, 2925, 3654, 4495, 5456};

// ---------------------------------------------------------------------------
// Kernel 1: S2 forward transform. One WG per (b, f_in) pair.
// fhat[b][s][i] = sum_j wig_s2[j][s] * ( sum_a x[b,i,j,a] e^{-2pi i m(s) a/64} )
// ---------------------------------------------------------------------------
__global__ void k1_fhat(const float* __restrict__ x,
                        const float* __restrict__ wig_s2,
                        float* __restrict__ fhat_r, float* __restrict__ fhat_i) {
  __shared__ float xs[64 * 64];
  __shared__ float xfr[64 * 32];
  __shared__ float xfi[64 * 32];
  __shared__ float ct[64];
  __shared__ float st[64];
  const int tid = threadIdx.x;
  const int bi  = blockIdx.x;  // b*16 + i

  const float* xp = x + (size_t)bi * 4096;
  for (int idx = tid; idx < 4096; idx += 256) xs[idx] = xp[idx];
  if (tid < 64) {
    float s, c;
    __sincosf(TWO_PI * (float)tid / 64.0f, &s, &c);
    ct[tid] = c; st[tid] = s;
  }
  __syncthreads();

  // DFT over alpha for m = -15..15 (31 bins), per beta row j
  for (int p = tid; p < 64 * 31; p += 256) {
    int j  = p / 31;
    int mi = p - j * 31;
    int m  = mi - 15;
    float ar = 0.f, ai = 0.f;
    const float* row = xs + j * 64;
    for (int a = 0; a < 64; ++a) {
      int t = (m * a) & 63;       // (m*a) mod 64, works for negative m
      float v = row[a];
      ar += v * ct[t];
      ai -= v * st[t];
    }
    xfr[j * 32 + mi] = ar;
    xfi[j * 32 + mi] = ai;
  }
  __syncthreads();

  // quadrature-weighted Wigner d_{m0} sum over beta; one spec coeff per thread
  {
    int s = tid;
    int l = 0;
    while ((l + 1) * (l + 1) <= s) ++l;
    int mi = (s - l * l - l) + 15;  // m + 15
    float fr = 0.f, fi = 0.f;
    for (int j = 0; j < 64; ++j) {
      float w = wig_s2[j * 256 + s];
      fr += w * xfr[j * 32 + mi];
      fi += w * xfi[j * 32 + mi];
    }
    int b = bi >> 4, i = bi & 15;
    int o = (b * 256 + s) * 16 + i;
    fhat_r[o] = fr;
    fhat_i[o] = fi;
  }
}

// ---------------------------------------------------------------------------
// Kernel 2: psi[s][i][o] = scaling * sum_g kernel[i,o,g] * Fk[s,g]
// ---------------------------------------------------------------------------
__global__ void k2_psi(const float* __restrict__ kern,
                       const float* __restrict__ fk_re,
                       const float* __restrict__ fk_im,
                       float* __restrict__ psi_r, float* __restrict__ psi_i,
                       float scaling) {
  int id = blockIdx.x * 256 + threadIdx.x;  // 262144 total
  int s   = id >> 10;
  int rem = id & 1023;
  int i   = rem >> 6;
  int o   = rem & 63;
  const float* kp = kern + (i * 64 + o) * 32;
  const float* fr = fk_re + s * 32;
  const float* fi = fk_im + s * 32;
  float pr = 0.f, pi = 0.f;
  for (int g = 0; g < 32; ++g) {
    float kv = kp[g];
    pr += kv * fr[g];
    pi += kv * fi[g];
  }
  int idx = (s * 16 + i) * 64 + o;
  psi_r[idx] = pr * scaling;
  psi_i[idx] = pi * scaling;
}

// ---------------------------------------------------------------------------
// Kernel 3: per-degree block outer product. One WG per (b, o).
// Fz[bo][t=(l,m,n)] = sum_i fhat[l,m,b,i] * conj(psi[l,n,i,o])
// ---------------------------------------------------------------------------
__global__ void k3_fz(const float* __restrict__ fhat_r, const float* __restrict__ fhat_i,
                      const float* __restrict__ psi_r,  const float* __restrict__ psi_i,
                      float* __restrict__ fz_r, float* __restrict__ fz_i) {
  __shared__ float fhr[256 * 16];
  __shared__ float fhi[256 * 16];
  const int tid = threadIdx.x;
  const int bo  = blockIdx.x;  // b*64 + o
  const int b = bo >> 6, o = bo & 63;
  const float* srcr = fhat_r + b * 4096;
  const float* srci = fhat_i + b * 4096;
  for (int idx = tid; idx < 4096; idx += 256) { fhr[idx] = srcr[idx]; fhi[idx] = srci[idx]; }
  __syncthreads();

  for (int t = tid; t < NSPEC3; t += 256) {
    int l = 15;
    while (c_off[l] > t) --l;
    int r = t - c_off[l];
    int L = 2 * l + 1;
    int mloc = r / L;
    int nloc = r - mloc * L;
    int sA = l * l + mloc;
    int sB = l * l + nloc;
    float zr = 0.f, zi = 0.f;
    const float* pr = psi_r + sB * 1024 + o;
    const float* pi = psi_i + sB * 1024 + o;
    for (int i = 0; i < 16; ++i) {
      float frv = fhr[sA * 16 + i];
      float fiv = fhi[sA * 16 + i];
      float prr = pr[i * 64];
      float pii = pi[i * 64];
      zr += frv * prr + fiv * pii;   // fhat * conj(psi)
      zi += fiv * prr - frv * pii;
    }
    fz_r[(size_t)bo * NSPEC3 + t] = zr;
    fz_i[(size_t)bo * NSPEC3 + t] = zi;
  }
}

// ---------------------------------------------------------------------------
// One 16x16 C-tile of a 32x32x32 f32 matmul via 8x V_WMMA_F32_16X16X4_F32.
// LDS pitch fixed at 33 floats (bank-conflict padding).
// ---------------------------------------------------------------------------
__device__ __forceinline__ v8f tile_mm(const float* A, int arow0, int acol0,
                                       const float* B, int brow0, int bcol0,
                                       float asign, float bsign, v8f c) {
  const int lane = threadIdx.x & 31;
  const int half = lane >> 4;     // 0: K pair (0,1); 1: K pair (2,3)
  const int l16  = lane & 15;
#pragma unroll
  for (int kb = 0; kb < 8; ++kb) {
    int ka = acol0 + kb * 4 + half * 2;
    v2f a, b;
    a.x = asign * A[(arow0 + l16) * 33 + ka];
    a.y = asign * A[(arow0 + l16) * 33 + ka + 1];
    int kr = brow0 + kb * 4 + half * 2;
    b.x = bsign * B[kr * 33 + bcol0 + l16];
    b.y = bsign * B[(kr + 1) * 33 + bcol0 + l16];
    c = __builtin_amdgcn_wmma_f32_16x16x4_f32(false, a, false, b, (short)0, c,
                                              false, false);
  }
  return c;
}

// ---------------------------------------------------------------------------
// Kernel 4: Wigner-d spec accumulation + inverse 2D DFT (WMMA).
// One WG (8 wave32) per (b, o, beta_k).
// ---------------------------------------------------------------------------
__global__ void k4_synth(const float* __restrict__ fz_r, const float* __restrict__ fz_i,
                         const float* __restrict__ wig_so3,
                         const float* __restrict__ bias,
                         float* __restrict__ out) {
  __shared__ float Sr[32 * 33], Si[32 * 33];
  __shared__ float Tr[32 * 33], Ti[32 * 33];
  __shared__ float Wr[32 * 33], Wi[32 * 33];
  const int tid = threadIdx.x;
  const int blk = blockIdx.x;
  const int k   = blk & 31;
  const int bo  = blk >> 5;
  const int o   = bo & 63;

  // twiddle table W[i][j] = e^{+2pi i*j/32} (symmetric; used for both axes)
  for (int idx = tid; idx < 1024; idx += 256) {
    int i = idx >> 5, j = idx & 31;
    float s, c;
    __sincosf(TWO_PI * (float)((i * j) & 31) / 32.0f, &s, &c);
    Wr[i * 33 + j] = c;
    Wi[i * 33 + j] = s;
  }

  // stage D: spec[m'][n'] = sum_{l>=max(|m|,|n|)} Fz^l[m,n] * d^l_{mn}(beta_k)
  const float* wk = wig_so3 + (size_t)k * NSPEC3;
  const float* fr = fz_r + (size_t)bo * NSPEC3;
  const float* fi = fz_i + (size_t)bo * NSPEC3;
  for (int cell = tid; cell < 1024; cell += 256) {
    int mi = cell >> 5, ni = cell & 31;
    int m = (mi < 16) ? mi : mi - 32;
    int n = (ni < 16) ? ni : ni - 32;
    int am = m < 0 ? -m : m;
    int an = n < 0 ? -n : n;
    float sr = 0.f, si = 0.f;
    if (am <= 15 && an <= 15) {
      int lmin = am > an ? am : an;
      for (int l = lmin; l <= 15; ++l) {
        int t = c_off[l] + (m + l) * (2 * l + 1) + (n + l);
        float w = wk[t];
        sr += fr[t] * w;
        si += fi[t] * w;
      }
    }
    Sr[mi * 33 + ni] = sr;
    Si[mi * 33 + ni] = si;
  }
  __syncthreads();

  // stage E1: T = S * E, E = Wr + i*Wi. 8 waves, one 16x16 tile each.
  {
    int w    = tid >> 5;     // wave id 0..7
    int part = w >> 2;       // 0 -> Tr, 1 -> Ti
    int ti   = (w >> 1) & 1;
    int tj   = w & 1;
    v8f c = {};
    if (part == 0) {         // Tr = Sr*Wr - Si*Wi
      c = tile_mm(Sr, ti * 16, 0, Wr, 0, tj * 16, 1.f,  1.f, c);
      c = tile_mm(Si, ti * 16, 0, Wi, 0, tj * 16, 1.f, -1.f, c);
    } else {                 // Ti = Sr*Wi + Si*Wr
      c = tile_mm(Sr, ti * 16, 0, Wi, 0, tj * 16, 1.f, 1.f, c);
      c = tile_mm(Si, ti * 16, 0, Wr, 0, tj * 16, 1.f, 1.f, c);
    }
    float* dst = (part == 0) ? Tr : Ti;
    const int lane = tid & 31;
    const int half = lane >> 4;
    const int l16  = lane & 15;
#pragma unroll
    for (int v = 0; v < 8; ++v)
      dst[(ti * 16 + v + half * 8) * 33 + tj * 16 + l16] = c[v];
  }
  __syncthreads();

  // stage E2: out = Re(G*T) = Wr*Tr - Wi*Ti. Waves 0..3, one tile each.
  {
    int w = tid >> 5;
    if (w < 4) {
      int ti = w >> 1, tj = w & 1;
      v8f c = {};
      c = tile_mm(Wr, ti * 16, 0, Tr, 0, tj * 16,  1.f, 1.f, c);
      c = tile_mm(Wi, ti * 16, 0, Ti, 0, tj * 16, -1.f, 1.f, c);
      const int lane = tid & 31;
      const int half = lane >> 4;
      const int l16  = lane & 15;
      float bv = bias[o];
      float* op = out + ((size_t)blk << 10);
#pragma unroll
      for (int v = 0; v < 8; ++v) {
        int alpha = ti * 16 + v + half * 8;
        int gamma = tj * 16 + l16;
        op[alpha * 32 + gamma] = c[v] + bv;
      }
    }
  }
}

// ---------------------------------------------------------------------------
extern "C" void kernel_launch(void* const* d_in, const int* in_sizes, int n_in,
                              void* d_out, int out_size, void* d_ws, size_t ws_size,
                              hipStream_t stream) {
  (void)in_sizes; (void)n_in; (void)out_size; (void)ws_size;
  const float* x       = (const float*)d_in[0];
  const float* kern    = (const float*)d_in[1];
  const float* bias    = (const float*)d_in[2];
  const float* wig_s2  = (const float*)d_in[3];
  const float* fk_re   = (const float*)d_in[4];
  const float* fk_im   = (const float*)d_in[5];
  const float* wig_so3 = (const float*)d_in[6];
  float* out = (float*)d_out;

  float* ws     = (float*)d_ws;
  float* fhat_r = ws;                       // 65536
  float* fhat_i = fhat_r + 65536;           // 65536
  float* psi_r  = fhat_i + 65536;           // 262144
  float* psi_i  = psi_r + 262144;           // 262144
  float* fz_r   = psi_i + 262144;           // 5456*1024
  float* fz_i   = fz_r + (size_t)NSPEC3 * 1024;

  const float scaling = 0.005524271728019903f;  // 1/sqrt(32*16*16^4/32^2)

  k1_fhat<<<256, 256, 0, stream>>>(x, wig_s2, fhat_r, fhat_i);
  k2_psi<<<1024, 256, 0, stream>>>(kern, fk_re, fk_im, psi_r, psi_i, scaling);
  k3_fz<<<1024, 256, 0, stream>>>(fhat_r, fhat_i, psi_r, psi_i, fz_r, fz_i);
  k4_synth<<<16 * 64 * 32, 256, 0, stream>>>(fz_r, fz_i, wig_so3, bias, out);
}